// RecurrentModel_49606872269494
// MI455X (gfx1250) — compile-verified
//
#include <hip/hip_runtime.h>
#include <math.h>
#include <stdint.h>

// ---------------------------------------------------------------------------
// y[b] = c0*x[b,0] + c1*x[b,1]   with
//   c0 = Q(1,1)+Q(1,2),  c1 = Q(2,1)-V(2,1)+V(2,2)   (row sums of Wc; c2==0)
// Memory bound: 128 MB traffic -> ~5.5us @ 23.3 TB/s. CDNA5 path used:
// async global->LDS DMA (ASYNCcnt) double buffered per wave, NT 128b stores.
// Stream kernel placed first in the file so its disasm is visible.
// ---------------------------------------------------------------------------

#define HQ 0.01  // H_INT

typedef float v4f __attribute__((ext_vector_type(4)));
typedef int   v4i __attribute__((ext_vector_type(4)));
typedef __attribute__((address_space(1))) v4i gv4i;   // global int4
typedef __attribute__((address_space(3))) v4i lv4i;   // LDS int4

#define ROWS_PER_TILE   128   // 32 lanes * 4 rows
#define FLOATS_PER_TILE 384   // 128 rows * 3 cols
#define WAVES_PER_BLOCK 8
#define BLOCK_THREADS   256
#define NUM_BLOCKS      2048

// ---- async copy of one 1536B wave tile (3 x b128 per lane) -----------------
__device__ __forceinline__ void rm_async_tile(const float* gsrc, float* lbuf, int lane) {
  gv4i* g = (gv4i*)(uintptr_t)(gsrc + lane * 4);
  lv4i* l = (lv4i*)(uint32_t)(uintptr_t)(lbuf + lane * 4);
  __builtin_amdgcn_global_load_async_to_lds_b128(g, l, 0,    0);
  __builtin_amdgcn_global_load_async_to_lds_b128(g, l, 512,  0);
  __builtin_amdgcn_global_load_async_to_lds_b128(g, l, 1024, 0);
}

// ---- streaming kernel ------------------------------------------------------
__global__ __launch_bounds__(BLOCK_THREADS)
void rm_stream_kernel(const float* __restrict__ x,
                      const float* __restrict__ coef,
                      float* __restrict__ y,
                      long long ntiles) {
  __shared__ __align__(16) float lds[2][WAVES_PER_BLOCK][FLOATS_PER_TILE];
  const int lane = threadIdx.x & 31;
  const int wib  = threadIdx.x >> 5;
  const long long wave_id = (long long)blockIdx.x * WAVES_PER_BLOCK + wib;
  const long long nwaves  = (long long)gridDim.x * WAVES_PER_BLOCK;

  const float c0 = coef[0];
  const float c1 = coef[1];

  float* buf0 = lds[0][wib];
  float* buf1 = lds[1][wib];

  long long tile = wave_id;
  if (tile < ntiles) rm_async_tile(x + tile * (long long)FLOATS_PER_TILE, buf0, lane);

  int cur = 0;
  for (; tile < ntiles; tile += nwaves) {
    const long long nxt = tile + nwaves;
    if (nxt < ntiles) {
      rm_async_tile(x + nxt * (long long)FLOATS_PER_TILE, cur ? buf0 : buf1, lane);
      __builtin_amdgcn_s_wait_asynccnt(3);   // current tile's 3 copies done
    } else {
      __builtin_amdgcn_s_wait_asynccnt(0);
    }
    const float* b = cur ? buf1 : buf0;
    const v4f* lp = (const v4f*)(b + lane * 12);   // 4 rows = 12 floats, 16B aligned
    const v4f f0 = lp[0];
    const v4f f1 = lp[1];
    const v4f f2 = lp[2];
    v4f r;
    r.x = c0 * f0.x + c1 * f0.y;   // row 0: x0,x1
    r.y = c0 * f0.w + c1 * f1.x;   // row 1
    r.z = c0 * f1.z + c1 * f1.w;   // row 2
    r.w = c0 * f2.y + c1 * f2.z;   // row 3
    __builtin_nontemporal_store(r, (v4f*)(y + tile * ROWS_PER_TILE + lane * 4));
    cur ^= 1;
  }
}

// ---- tail (B % 128 rows; 0 for B = 8M, kept for generality) ----------------
__global__ void rm_tail_kernel(const float* __restrict__ x,
                               const float* __restrict__ coef,
                               float* __restrict__ y,
                               long long startRow, long long B) {
  long long r = startRow + (long long)blockIdx.x * blockDim.x + threadIdx.x;
  if (r < B) y[r] = coef[0] * x[r * 3] + coef[1] * x[r * 3 + 1];
}

// ---- tiny quadrature kernel: computes c0,c1 into ws[0..1] ------------------
__global__ void rm_coef_kernel(const float* __restrict__ s,
                               const float* __restrict__ w,
                               const float* __restrict__ p,
                               float* __restrict__ coef) {
  __shared__ double part[100][5];
  const int k = threadIdx.x;
  if (k < 100) {
    const double t  = (double)k / 99.0;          // linspace(n,n+1,100) - n
    const double s0 = s[0], s1 = s[1], s2 = s[2];
    const double w00 = w[0], w01 = w[1], w10 = w[2], w11 = w[3];
    const double p0 = p[0], p1 = p[1];
    auto kern = [&](double u, double v) {
      return s0 + HQ * (s1 * cos(u * w00 + v * w01 - p0) +
                        s2 * cos(u * w10 + v * w11 - p1));
    };
    const double u1 = 2.0 + t;   // n=1: u = n + i = 2n + t
    const double u2 = 4.0 + t;   // n=2
    const double A  = kern(u1, 1.0);              // Q(1,1) integrand
    const double Bq = kern(u2, 1.0);              // Q(1,2) integrand
    const double Cq = kern(u1, 2.0);              // Q(2,1) integrand
    const double Dv = Cq * (1.0 + t);             // V(2,1) integrand (i = n+t)
    const double Ev = kern(u2, 2.0) * (2.0 + t);  // V(2,2) integrand
    part[k][0] = A; part[k][1] = Bq; part[k][2] = Cq; part[k][3] = Dv; part[k][4] = Ev;
  }
  __syncthreads();
  if (k == 0) {
    double S0 = 0, S1 = 0, S2 = 0, S3 = 0, S4 = 0;
    for (int i = 0; i < 100; ++i) {   // fixed order: deterministic
      S0 += part[i][0]; S1 += part[i][1]; S2 += part[i][2];
      S3 += part[i][3]; S4 += part[i][4];
    }
    coef[0] = (float)(HQ * S0 + HQ * S1);           // c0
    coef[1] = (float)(HQ * S2 - HQ * S3 + HQ * S4); // c1
  }
}

extern "C" void kernel_launch(void* const* d_in, const int* in_sizes, int n_in,
                              void* d_out, int out_size, void* d_ws, size_t ws_size,
                              hipStream_t stream) {
  const float* x = (const float*)d_in[0];
  const float* s = (const float*)d_in[1];
  const float* w = (const float*)d_in[2];
  const float* p = (const float*)d_in[3];
  float* y    = (float*)d_out;
  float* coef = (float*)d_ws;

  const long long B      = (long long)in_sizes[0] / 3;
  const long long ntiles = B / ROWS_PER_TILE;
  const long long rem    = B - ntiles * ROWS_PER_TILE;

  rm_coef_kernel<<<1, 128, 0, stream>>>(s, w, p, coef);
  rm_stream_kernel<<<NUM_BLOCKS, BLOCK_THREADS, 0, stream>>>(x, coef, y, ntiles);
  if (rem > 0) {
    const long long start = ntiles * ROWS_PER_TILE;
    const int nb = (int)((rem + 255) / 256);
    rm_tail_kernel<<<nb, 256, 0, stream>>>(x, coef, y, start, B);
  }
}